// MaxPoolStack_19782619365608
// MI455X (gfx1250) — compile-verified
//
#include <hip/hip_runtime.h>
#include <cstdint>

// Problem constants (from reference): STACK=2, N=32, I=4096, D=128, K=16
#define STACK   2
#define NPC     32       // point clouds
#define IPTS    4096     // points per cloud
#define DFEA    128      // features
#define KSEG    16       // clusters per cloud
#define CHUNKS  16       // split I across blocks for occupancy
#define ROWS_PER_CHUNK (IPTS / CHUNKS)   // 256
#define TPB     256      // 8 waves (wave32)

typedef float v4f __attribute__((ext_vector_type(4)));

__global__ __launch_bounds__(TPB) void segmax_init_kernel(float* __restrict__ out, int n) {
    int i = blockIdx.x * blockDim.x + threadIdx.x;
    if (i < n) out[i] = -__builtin_inff();   // segment_max identity (matches JAX empty-segment)
}

__global__ __launch_bounds__(TPB) void segmax_kernel(const float* __restrict__ feat,
                                                     const int*   __restrict__ cidx,
                                                     float*       __restrict__ out) {
    __shared__ float acc[KSEG * DFEA];        // 8 KB partial-max accumulator
    __shared__ int   idxbuf[ROWS_PER_CHUNK];  // 1 KB staged cluster indices

    const int t     = threadIdx.x;
    const int chunk = blockIdx.x;
    const int n     = blockIdx.y;
    const int s     = blockIdx.z;

    // ---- init LDS accumulator to -inf ----
    #pragma unroll
    for (int j = 0; j < (KSEG * DFEA) / TPB; ++j)
        acc[j * TPB + t] = -__builtin_inff();

    // ---- async-copy this chunk's 256 int32 cluster ids into LDS (ASYNCcnt path) ----
    {
        const int* gsrc = cidx + ((size_t)(s * NPC + n) * IPTS + (size_t)chunk * ROWS_PER_CHUNK + t);
        unsigned lds_addr = (unsigned)(uintptr_t)(&idxbuf[t]);  // low 32b of generic ptr = LDS offset
        asm volatile("global_load_async_to_lds_b32 %0, %1, off"
                     :: "v"(lds_addr), "v"(gsrc) : "memory");
    }
    asm volatile("s_wait_asynccnt 0" ::: "memory");
    __syncthreads();

    const int warp = t >> 5;          // wave id 0..7 (wave32)
    const int lane = t & 31;
    const int d4   = lane << 2;       // each lane owns one float4 of a row -> full 512B row per wave

    const float* fbase = feat + ((size_t)(s * NPC + n) * IPTS + (size_t)chunk * ROWS_PER_CHUNK) * DFEA;

    // 256 rows per chunk, 8 waves, 4 rows per wave per iteration -> 8 iterations.
    #pragma unroll 1
    for (int it = 0; it < ROWS_PER_CHUNK; it += 32) {
        v4f      v[4];
        unsigned a[4];
        #pragma unroll
        for (int u = 0; u < 4; ++u) {
            const int r = it + warp + u * 8;
            // non-temporal streaming load: data is touched exactly once
            v[u] = __builtin_nontemporal_load((const v4f*)(fbase + (size_t)r * DFEA + d4));
            const int k = idxbuf[r] - n * KSEG;               // local cluster id in [0,16)
            a[u] = (unsigned)(uintptr_t)(&acc[k * DFEA + d4]);
        }
        #pragma unroll
        for (int u = 0; u < 4; ++u) {
            // wave-uniform row => wave-uniform k => 32 lanes hit 32 distinct banks (conflict-free)
            asm volatile("ds_max_num_f32 %0, %1\n\t"
                         "ds_max_num_f32 %0, %2 offset:4\n\t"
                         "ds_max_num_f32 %0, %3 offset:8\n\t"
                         "ds_max_num_f32 %0, %4 offset:12"
                         :: "v"(a[u]), "v"(v[u].x), "v"(v[u].y), "v"(v[u].z), "v"(v[u].w)
                         : "memory");
        }
    }

    asm volatile("s_wait_dscnt 0x0" ::: "memory");  // asm ds atomics are untracked by compiler
    __syncthreads();

    // ---- merge 16x128 partial maxima into global output (float atomic max = deterministic) ----
    float* obase = out + (size_t)(s * NPC + n) * (KSEG * DFEA);
    #pragma unroll
    for (int j = 0; j < (KSEG * DFEA) / TPB; ++j) {
        const int    slot = j * TPB + t;
        const float  vv   = acc[slot];
        const float* gp   = obase + slot;
        asm volatile("global_atomic_max_num_f32 %0, %1, off"
                     :: "v"(gp), "v"(vv) : "memory");
    }
    // s_endpgm performs implicit wait-idle, covering the outstanding atomics.
}

extern "C" void kernel_launch(void* const* d_in, const int* in_sizes, int n_in,
                              void* d_out, int out_size, void* d_ws, size_t ws_size,
                              hipStream_t stream) {
    const float* feat = (const float*)d_in[0];   // [2,32,4096,128] f32
    const int*   idx  = (const int*)d_in[1];     // [2,131072] int (harness delivers 32-bit ints)
    float*       out  = (float*)d_out;           // [2,32,16,128] f32

    // Pass 1: seed output with -inf (d_out is poisoned by harness; must re-init every call)
    segmax_init_kernel<<<(out_size + TPB - 1) / TPB, TPB, 0, stream>>>(out, out_size);

    // Pass 2: chunked segment-max with LDS ds_max_num_f32 + global_atomic_max_num_f32 merge
    dim3 grid(CHUNKS, NPC, STACK);   // 16 x 32 x 2 = 1024 blocks
    segmax_kernel<<<grid, TPB, 0, stream>>>(feat, idx, out);
}